// flash_decode_layer_rccl_33749853012678
// MI455X (gfx1250) — compile-verified
//
#include <hip/hip_runtime.h>
#include <math.h>

// Problem constants (match reference).
#define NRANK 4
#define BATCH 16
#define NHQ 32
#define NHKV 8
#define GRP 4                 // NHQ / NHKV
#define DQK 128
#define DV 128
#define NUM_PAGES 4096
#define PAGE_SZ 16
#define PAGES_PER_SEQ 128

// Split-S decode: each wave handles one (rank, batch, kv-head, split) partial.
#define SPLIT 8
#define PAGES_PER_SPLIT (PAGES_PER_SEQ / SPLIT)   // 16 pages = 256 keys
#define NPART (NRANK * SPLIT)                     // 32 partials per (b, hq)
#define WAVES_PER_BLOCK 4

typedef float v2f __attribute__((ext_vector_type(2)));
typedef float v8f __attribute__((ext_vector_type(8)));

static constexpr float kScale  = 0.08838834764831845f;
static constexpr float kNegMax = 3.402823466e38f;

// ---------------------------------------------------------------------------
// Pass 1: one wave32 per (r, b, kvh, split).
// Scores computed transposed:  Sᵀ(16 keys × 16 qcols) = K(16x4) x Qᵀ(4x16)
// Output accumulated transposed: outᵀ(vd × qcols)    += Vᵀ(16x4) x eᵀ(4x16)
// using V_WMMA_F32_16X16X4_F32 chains along the 128-dim / 16-key axes.
// ---------------------------------------------------------------------------
__global__ __launch_bounds__(32 * WAVES_PER_BLOCK)
void fd_pass1(const float* __restrict__ q,
              const float* __restrict__ k_cache,
              const float* __restrict__ v_cache,
              const int*   __restrict__ kv_lens,
              const int*   __restrict__ block_table,
              float*       __restrict__ out_ws,   // [NPART][BATCH][NHQ][DV]
              float*       __restrict__ lse_ws)   // [NPART][BATCH][NHQ]
{
    const int lane = threadIdx.x & 31;
    const int wid  = threadIdx.x >> 5;
    int gid = blockIdx.x * WAVES_PER_BLOCK + wid; // [0, NRANK*BATCH*NHKV*SPLIT)
    const int split = gid % SPLIT;  gid /= SPLIT;
    const int kvh   = gid % NHKV;   gid /= NHKV;
    const int b     = gid % BATCH;
    const int r     = gid / BATCH;

    const int m    = lane & 15;   // key row (A) == query column (B) index
    const int half = lane >> 4;   // lane half selects K-chunk sub-pair

    // ---- Q fragments (B operand; rows >= GRP zeroed; pre-scaled) ----------
    v2f qf[32];
    {
        const float* qrow =
            q + ((size_t)b * NHQ + kvh * GRP + (m < GRP ? m : 0)) * DQK;
        #pragma unroll
        for (int c = 0; c < 32; ++c) {
            if (m < GRP) {
                const float* p2 = qrow + 4 * c + 2 * half;
                qf[c][0] = p2[0] * kScale;
                qf[c][1] = p2[1] * kScale;
            } else {
                qf[c][0] = 0.f;
                qf[c][1] = 0.f;
            }
        }
    }

    const int kv_len     = kv_lens[r * BATCH + b];
    const int split_base = split * (PAGES_PER_SPLIT * PAGE_SZ);
    int rem    = kv_len - split_base;
    int npages = (rem <= 0) ? 0 : (rem + PAGE_SZ - 1) / PAGE_SZ;
    if (npages > PAGES_PER_SPLIT) npages = PAGES_PER_SPLIT;

    const int* bt = block_table + ((size_t)r * BATCH + b) * PAGES_PER_SEQ
                                + split * PAGES_PER_SPLIT;

    v8f acc[8];
    #pragma unroll
    for (int t = 0; t < 8; ++t) {
        #pragma unroll
        for (int j = 0; j < 8; ++j) acc[t][j] = 0.f;
    }
    float m_run = -kNegMax;
    float l_run = 0.f;

    for (int p = 0; p < npages; ++p) {
        const int    page_id = bt[p];
        const size_t pbase   = ((size_t)r * NUM_PAGES + page_id) * PAGE_SZ;
        const float* krow    = k_cache + ((pbase + m) * NHKV + kvh) * DQK;
        const float* vrow0   = v_cache + (pbase * NHKV + kvh) * DV;

        // Prefetch next page's K/V rows (global_prefetch_b8).
        if (p + 1 < npages) {
            const size_t nb = ((size_t)r * NUM_PAGES + bt[p + 1]) * PAGE_SZ;
            const float* nk = k_cache + ((nb + m) * NHKV + kvh) * DQK;
            const float* nv = v_cache + ((nb + m) * NHKV + kvh) * DV;
            __builtin_prefetch(nk + 64 * half, 0, 3);
            __builtin_prefetch(nk + 64 * half + 32, 0, 3);
            __builtin_prefetch(nv + 64 * half, 0, 3);
            __builtin_prefetch(nv + 64 * half + 32, 0, 3);
        }

        // ---- K fragments + Sᵀ = K · Qᵀ  (32 x WMMA f32 16x16x4) ----------
        v2f kf[32];
        #pragma unroll
        for (int c = 0; c < 32; ++c) {
            const float* p2 = krow + 4 * c + 2 * half;
            kf[c][0] = p2[0];
            kf[c][1] = p2[1];
        }
        v8f s0, s1;
        #pragma unroll
        for (int j = 0; j < 8; ++j) { s0[j] = 0.f; s1[j] = 0.f; }
        #pragma unroll
        for (int c = 0; c < 32; c += 2) {
            s0 = __builtin_amdgcn_wmma_f32_16x16x4_f32(
                     false, kf[c],     false, qf[c],     (short)0, s0, false, false);
            s1 = __builtin_amdgcn_wmma_f32_16x16x4_f32(
                     false, kf[c + 1], false, qf[c + 1], (short)0, s1, false, false);
        }
        const v8f s = s0 + s1;

        // ---- online softmax; this lane's keys are (kbase + j) -------------
        const int kbase = split_base + p * PAGE_SZ + 8 * half;
        float pj[8];
        float smax = -kNegMax;
        #pragma unroll
        for (int j = 0; j < 8; ++j) {
            const float sj = (kbase + j) < kv_len ? s[j] : -kNegMax;
            pj[j] = sj;
            smax  = fmaxf(smax, sj);
        }
        smax = fmaxf(smax, __shfl_xor(smax, 16, 32));
        const float m_new = fmaxf(m_run, smax);
        const float alpha = __expf(m_run - m_new);
        float psum = 0.f;
        #pragma unroll
        for (int j = 0; j < 8; ++j) {
            const float e = (kbase + j) < kv_len ? __expf(pj[j] - m_new) : 0.f;
            pj[j] = e;
            psum += e;
        }
        psum += __shfl_xor(psum, 16, 32);
        l_run = l_run * alpha + psum;
        m_run = m_new;

        #pragma unroll
        for (int t = 0; t < 8; ++t) {
            #pragma unroll
            for (int j = 0; j < 8; ++j) acc[t][j] *= alpha;
        }

        // Partner half's 8 probabilities (keys of the other lane half).
        float qj[8];
        #pragma unroll
        for (int j = 0; j < 8; ++j) qj[j] = __shfl_xor(pj[j], 16, 32);

        // eᵀ B-fragments, chunk c2 covers keys {4c2+2*half, +1}.
        v2f ef[4];
        ef[0][0] = half ? qj[2] : pj[0];  ef[0][1] = half ? qj[3] : pj[1];
        ef[1][0] = half ? qj[6] : pj[4];  ef[1][1] = half ? qj[7] : pj[5];
        ef[2][0] = half ? pj[2] : qj[0];  ef[2][1] = half ? pj[3] : qj[1];
        ef[3][0] = half ? pj[6] : qj[4];  ef[3][1] = half ? pj[7] : qj[5];

        // ---- outᵀ += Vᵀ · eᵀ  (8 vd-tiles x 4 key-chunks) -----------------
        #pragma unroll
        for (int c2 = 0; c2 < 4; ++c2) {
            const int    k0 = 4 * c2 + 2 * half;
            const float* va = vrow0 + (size_t)k0 * (NHKV * DV);
            const float* vb = va + (NHKV * DV);
            #pragma unroll
            for (int t = 0; t < 8; ++t) {
                v2f vf;
                vf[0] = va[16 * t + m];
                vf[1] = vb[16 * t + m];
                acc[t] = __builtin_amdgcn_wmma_f32_16x16x4_f32(
                             false, vf, false, ef[c2], (short)0, acc[t],
                             false, false);
            }
        }
    }

    // ---- emit normalized partial + lse ------------------------------------
    if (m < GRP) {
        const size_t pidx = (size_t)(r * SPLIT + split) * BATCH + b;
        const int    hq   = kvh * GRP + m;
        const float  inv  = 1.f / fmaxf(l_run, 1e-38f);
        float* op = out_ws + (pidx * NHQ + hq) * DV;
        #pragma unroll
        for (int t = 0; t < 8; ++t) {
            #pragma unroll
            for (int j = 0; j < 8; ++j)
                op[16 * t + 8 * half + j] = acc[t][j] * inv;  // vd = 16t+8*half+j
        }
        if (half == 0) {
            lse_ws[pidx * NHQ + hq] =
                (l_run > 0.f) ? (m_run + __logf(l_run)) : -kNegMax;
        }
    }
}

// ---------------------------------------------------------------------------
// Pass 2: flat LSE-weighted combine of the 32 partials per (b, hq).
// ---------------------------------------------------------------------------
__global__ __launch_bounds__(DV)
void fd_pass2(const float* __restrict__ out_ws,
              const float* __restrict__ lse_ws,
              float*       __restrict__ out)
{
    const int bh = blockIdx.x;        // b * NHQ + hq
    const int vd = threadIdx.x;       // 0..DV-1

    float l[NPART];
    float lmax = -kNegMax;
    #pragma unroll
    for (int p = 0; p < NPART; ++p) {
        l[p] = lse_ws[(size_t)p * (BATCH * NHQ) + bh];
        lmax = fmaxf(lmax, l[p]);
    }
    float wsum = 0.f, a = 0.f;
    #pragma unroll
    for (int p = 0; p < NPART; ++p) {
        const float w = __expf(l[p] - lmax);
        wsum += w;
        a += w * out_ws[((size_t)p * (BATCH * NHQ) + bh) * DV + vd];
    }
    out[(size_t)bh * DV + vd] = a / wsum;
}

// ---------------------------------------------------------------------------
extern "C" void kernel_launch(void* const* d_in, const int* in_sizes, int n_in,
                              void* d_out, int out_size, void* d_ws,
                              size_t ws_size, hipStream_t stream)
{
    const float* q           = (const float*)d_in[0];
    const float* k_cache     = (const float*)d_in[1];
    const float* v_cache     = (const float*)d_in[2];
    const int*   kv_lens     = (const int*)d_in[3];
    const int*   block_table = (const int*)d_in[4];
    float*       out         = (float*)d_out;

    float* out_ws = (float*)d_ws;                                  // 8 MB
    float* lse_ws = out_ws + (size_t)NPART * BATCH * NHQ * DV;     // 64 KB

    const int waves = NRANK * BATCH * NHKV * SPLIT;                // 4096
    fd_pass1<<<waves / WAVES_PER_BLOCK, 32 * WAVES_PER_BLOCK, 0, stream>>>(
        q, k_cache, v_cache, kv_lens, block_table, out_ws, lse_ws);

    fd_pass2<<<BATCH * NHQ, DV, 0, stream>>>(out_ws, lse_ws, out);
}